// GlobalBundleAdjustment_15427522527437
// MI455X (gfx1250) — compile-verified
//
#include <hip/hip_runtime.h>
#include <math.h>

#define N_VIEWS   512
#define N_MULTI   32
#define MAX_SIFT  2048
#define N_MATCHES 1000000
#define FEAT_ELEMS ((size_t)N_VIEWS * MAX_SIFT * 3)

typedef __attribute__((ext_vector_type(2))) float v2f;
typedef __attribute__((ext_vector_type(8))) float v8f;

// ---------------------------------------------------------------------------
// Kernel 1: Euler -> rotation, Hillis-Steele scan of cumulative matrix product
// (non-commutative, left operand = earlier prefix) and cumsum of translations.
// One block of 512 threads; 9 scan steps.
// ---------------------------------------------------------------------------
__global__ void ba_scan_kernel(const float* __restrict__ euler,
                               const float* __restrict__ c3in,
                               float* __restrict__ c33ws,   // [512*9]
                               float* __restrict__ c3ws) {  // [512*3]
    __shared__ float Ms[N_VIEWS * 9];
    __shared__ float Ts[N_VIEWS * 3];
    const int t = threadIdx.x;

    const float ea = euler[t * 3 + 0];
    const float eb = euler[t * 3 + 1];
    const float ec = euler[t * 3 + 2];
    const float ca = cosf(ea), cb = cosf(eb), cc = cosf(ec);
    const float sa = sinf(ea), sb = sinf(eb), sc = sinf(ec);

    float M[9];
    M[0] = ca * cc + sa * sb * sc;  M[1] = -sa * cb;  M[2] = -ca * sc + sa * sb * cc;
    M[3] = sa * cc - ca * sb * sc;  M[4] =  ca * cb;  M[5] = -sa * sc - ca * sb * cc;
    M[6] = cb * sc;                 M[7] =  sb;       M[8] =  cb * cc;

    #pragma unroll
    for (int i = 0; i < 9; ++i) Ms[t * 9 + i] = M[i];
    #pragma unroll
    for (int i = 0; i < 3; ++i) Ts[t * 3 + i] = c3in[t * 3 + i];
    __syncthreads();

    for (int d = 1; d < N_VIEWS; d <<= 1) {
        float L[9], R[9], Lt[3], Rt[3];
        const bool act = (t >= d);
        if (act) {
            #pragma unroll
            for (int i = 0; i < 9; ++i) { L[i] = Ms[(t - d) * 9 + i]; R[i] = Ms[t * 9 + i]; }
            #pragma unroll
            for (int i = 0; i < 3; ++i) { Lt[i] = Ts[(t - d) * 3 + i]; Rt[i] = Ts[t * 3 + i]; }
        }
        __syncthreads();
        if (act) {
            #pragma unroll
            for (int r = 0; r < 3; ++r)
                #pragma unroll
                for (int j = 0; j < 3; ++j)
                    Ms[t * 9 + r * 3 + j] =
                        L[r * 3 + 0] * R[0 * 3 + j] +
                        L[r * 3 + 1] * R[1 * 3 + j] +
                        L[r * 3 + 2] * R[2 * 3 + j];
            #pragma unroll
            for (int i = 0; i < 3; ++i) Ts[t * 3 + i] = Lt[i] + Rt[i];
        }
        __syncthreads();
    }

    #pragma unroll
    for (int i = 0; i < 9; ++i) c33ws[t * 9 + i] = Ms[t * 9 + i];
    #pragma unroll
    for (int i = 0; i < 3; ++i) c3ws[t * 3 + i] = Ts[t * 3 + i];
}

// ---------------------------------------------------------------------------
// Kernel 2: fused (beta-contraction + rotation + translation) as a per-view
// skinny GEMM: D = W(3x99) x P(99x2048) via V_WMMA_F32_16X16X4_F32.
//   W[j, c*32+m] = beta[v,m] * c33[v][c][j];  W[j, 96+c] = c33[v][c][j]
//   P rows 0..95 = pdepth_xyz[v] (K-major == memory order), 96..98 = pdepth0.
// Grid (512, 16), 256 threads (8 waves x 16 s-columns).
// ---------------------------------------------------------------------------
__global__ void ba_feat_kernel(const float* __restrict__ beta,
                               const float* __restrict__ pdepth,
                               const float* __restrict__ pdepth0,
                               const float* __restrict__ c33ws,
                               const float* __restrict__ c3ws,
                               float* __restrict__ feat_out) {
    __shared__ float Wlds[100 * 16]; // [k][j], k<100 (k==99 zero pad), j<16 (j>=3 zero)
    __shared__ float c3s[3];

    const int v = blockIdx.x;

    // Stage W into LDS (1600 entries, 256 threads).
    for (int e = threadIdx.x; e < 1600; e += 256) {
        const int k = e >> 4;
        const int j = e & 15;
        float val = 0.0f;
        if (j < 3 && k < 99) {
            if (k < 96)
                val = beta[v * N_MULTI + (k & 31)] * c33ws[v * 9 + (k >> 5) * 3 + j];
            else
                val = c33ws[v * 9 + (k - 96) * 3 + j];
        }
        Wlds[e] = val;
    }
    if (threadIdx.x < 3) c3s[threadIdx.x] = c3ws[v * 3 + threadIdx.x];
    __syncthreads();

    const int lane = threadIdx.x & 31;
    const int wv   = threadIdx.x >> 5;
    const int half = lane >> 4;   // 0: K pair {k0,k0+1}; 1: {k0+2,k0+3}
    const int q    = lane & 15;   // A: M-row (=j); B: N-col (=s_local)
    const int sb   = blockIdx.y * 128 + wv * 16;
    const int col  = sb + q;

    const float* __restrict__ Pv  = pdepth  + (size_t)v * 96 * MAX_SIFT;
    const float* __restrict__ P0v = pdepth0 + (size_t)v * 3  * MAX_SIFT;

    v8f acc = {};

    // 24 steps over K = 0..95 (pdepth rows, memory-contiguous per row).
    #pragma unroll 4
    for (int t = 0; t < 24; ++t) {
        const int kk = 4 * t + 2 * half;
        v2f a, b;
        a.x = Wlds[kk * 16 + q];
        a.y = Wlds[(kk + 1) * 16 + q];
        b.x = Pv[(size_t)kk * MAX_SIFT + col];
        b.y = Pv[(size_t)(kk + 1) * MAX_SIFT + col];
        acc = __builtin_amdgcn_wmma_f32_16x16x4_f32(false, a, false, b,
                                                    (short)0, acc, false, false);
    }

    // Tail: K = 96..98 from pdepth0, K = 99 zero-padded.
    {
        const int kk = 96 + 2 * half;
        v2f a, b;
        a.x = Wlds[kk * 16 + q];
        a.y = Wlds[(kk + 1) * 16 + q];
        b.x = P0v[(size_t)(kk - 96) * MAX_SIFT + col];
        if (half == 0)
            b.y = P0v[(size_t)1 * MAX_SIFT + col];
        else
            b.y = 0.0f;
        acc = __builtin_amdgcn_wmma_f32_16x16x4_f32(false, a, false, b,
                                                    (short)0, acc, false, false);
    }

    // D layout: vgpr r, lanes 0-15 -> M=r (=j), N=lane (=s_local).
    // Only M=0..2 are real outputs; lanes 0-15 each write 3 contiguous floats.
    if (lane < 16) {
        const int s = sb + lane;
        float* o = feat_out + ((size_t)v * MAX_SIFT + s) * 3;
        o[0] = acc[0] + c3s[0];
        o[1] = acc[1] + c3s[1];
        o[2] = acc[2] + c3s[2];
    }
}

// ---------------------------------------------------------------------------
// Kernel 3: match distances (gathers hit the 12.6 MB feat array, L2-resident).
// ---------------------------------------------------------------------------
__global__ void ba_delta_kernel(const int* __restrict__ matches,
                                const float* __restrict__ feat,
                                float* __restrict__ delta,
                                int n) {
    const int i = blockIdx.x * blockDim.x + threadIdx.x;
    if (i >= n) return;
    const int i0 = matches[2 * i + 0];
    const int i1 = matches[2 * i + 1];
    const float* p0 = feat + (size_t)i0 * 3;
    const float* p1 = feat + (size_t)i1 * 3;
    const float dx = p0[0] - p1[0];
    const float dy = p0[1] - p1[1];
    const float dz = p0[2] - p1[2];
    delta[i] = sqrtf(dx * dx + dy * dy + dz * dz);
}

// ---------------------------------------------------------------------------
extern "C" void kernel_launch(void* const* d_in, const int* in_sizes, int n_in,
                              void* d_out, int out_size, void* d_ws, size_t ws_size,
                              hipStream_t stream) {
    const float* euler   = (const float*)d_in[0]; // (512,3)
    const float* c3in    = (const float*)d_in[1]; // (512,1,3)
    const float* beta    = (const float*)d_in[2]; // (512,1,1,32)
    const float* pdepth  = (const float*)d_in[3]; // (512,3,32,2048)
    const float* pdepth0 = (const float*)d_in[4]; // (512,3,1,2048)
    const int*   matches = (const int*)d_in[5];   // (1000000,2)

    float* out  = (float*)d_out;                  // feat (512*2048*3) ++ delta (1e6)
    float* c33ws = (float*)d_ws;                  // 512*9 floats
    float* c3ws  = c33ws + N_VIEWS * 9;           // 512*3 floats

    ba_scan_kernel<<<1, N_VIEWS, 0, stream>>>(euler, c3in, c33ws, c3ws);

    ba_feat_kernel<<<dim3(N_VIEWS, MAX_SIFT / 128), 256, 0, stream>>>(
        beta, pdepth, pdepth0, c33ws, c3ws, out);

    ba_delta_kernel<<<(N_MATCHES + 255) / 256, 256, 0, stream>>>(
        matches, out, out + FEAT_ELEMS, N_MATCHES);
}